// ThorMoE_52304111730967
// MI455X (gfx1250) — compile-verified
//
#include <hip/hip_runtime.h>
#include <math.h>
#include <stdint.h>

#define E_   8
#define H_   1024
#define I_   4096
#define NTOK 8192

typedef __attribute__((ext_vector_type(2))) float v2f;
typedef __attribute__((ext_vector_type(8))) float v8f;

__device__ __forceinline__ v8f wmma_f32(v2f a, v2f b, v8f c) {
  // D = A(16x4) * B(4x16) + C(16x16), fp32 WMMA (CDNA5)
  return __builtin_amdgcn_wmma_f32_16x16x4_f32(
      /*neg_a=*/false, a, /*neg_b=*/false, b,
      /*c_mod=*/(short)0, c, /*reuse_a=*/false, /*reuse_b=*/false);
}

__device__ __forceinline__ float gelu_exact(float v) {
  return 0.5f * v * (1.0f + erff(v * 0.70710678118654752440f));
}

// Async copy: memory -> LDS, 16 bytes per lane, tracked with ASYNCcnt.
__device__ __forceinline__ void async_b128(uint32_t lds_off, uint64_t gaddr) {
  asm volatile("global_load_async_to_lds_b128 %0, %1, off"
               :: "v"(lds_off), "v"(gaddr) : "memory");
}
__device__ __forceinline__ void wait_async0() {
  asm volatile("s_wait_asynccnt 0x0" ::: "memory");
}

// ---------------------------------------------------------------- routing ---
__global__ void zero_counts_kernel(int* __restrict__ counts) {
  if (threadIdx.x < E_) counts[threadIdx.x] = 0;
}

__global__ void route_kernel(const int* __restrict__ expert_idx,
                             int* __restrict__ counts,
                             int* __restrict__ buckets) {
  int t = blockIdx.x * blockDim.x + threadIdx.x;
  if (t < NTOK) {
    int e = expert_idx[t];
    int p = atomicAdd(&counts[e], 1);
    buckets[e * NTOK + p] = t;
  }
}

// ------------------------------------------------------------- fused GEMM ---
// Out[64x128 tile] = gather(Ag)[64xK_] * W[e][K_ x 128-slice] (+bias)
// GELU_=true : epilogue applies exact GELU, scatters to Out[token][LDN_]
// GELU_=false: epilogue adds residual Xres[token][LDN_], scatters to Out.
// A tiles (gathered rows) and W tiles are staged in LDS with
// GLOBAL_LOAD_ASYNC_TO_LDS_B128 (ASYNCcnt), double buffered.
template <int K_, int LDA_, int LDN_, bool GELU_>
__global__ __launch_bounds__(256) void moe_gemm(
    const float* __restrict__ Ag, const float* __restrict__ Wg,
    const float* __restrict__ bias_g, const float* __restrict__ Xres,
    const int* __restrict__ counts, const int* __restrict__ buckets,
    float* __restrict__ Out) {
  const int e   = blockIdx.z;
  const int cnt = counts[e];
  const int m0  = blockIdx.y * 64;
  if (m0 >= cnt) return;
  const int n0 = blockIdx.x * 128;
  const float* __restrict__ W    = Wg + (size_t)e * K_ * LDN_;
  const float* __restrict__ bias = bias_g + e * LDN_;

  __shared__ __align__(16) float sA[2][64 * 20];    // A: stride 20 (bank-safe)
  __shared__ __align__(16) float sB[2][16 * 136];   // B: stride 136 (bank-safe)
  __shared__ int sTok[64];

  const int t = threadIdx.x;
  if (t < 64) {
    int r = m0 + t;
    // clamp instead of branch: keeps EXEC all-ones in the main loop; rows
    // beyond cnt compute garbage that the guarded epilogue discards.
    sTok[t] = buckets[e * NTOK + ((r < cnt) ? r : (cnt - 1))];
  }
  __syncthreads();

  const int lane = t & 31, wave = t >> 5;
  const int l16  = lane & 15, half = lane >> 4, klo = half * 2;
  const int wm   = (wave >> 2) * 32;   // 2 wave rows  (64/32)
  const int wn   = (wave & 3) * 32;    // 4 wave cols (128/32)

  // ---- async loader mapping (each wave moves 512B per instruction) ----
  // A tile 64x16 f32: wave w covers rows 8w..8w+7, 4 lanes per 64B row.
  const int arow = wave * 8 + (lane >> 2);
  const int acb  = (lane & 3) * 16;            // byte offset within A row
  // B tile 16x128 f32: wave w covers rows 2w, 2w+1; 1 lane per 16B chunk.
  const int brow0 = wave * 2, brow1 = wave * 2 + 1;
  const int bcb   = lane * 16;                 // byte offset within B row

  const int tokA = sTok[arow];
  uint64_t gA  = (uint64_t)(uintptr_t)Ag + ((uint64_t)tokA * LDA_) * 4ull + (uint64_t)acb;
  uint64_t gB0 = (uint64_t)(uintptr_t)(W + (size_t)brow0 * LDN_ + n0) + (uint64_t)bcb;
  uint64_t gB1 = (uint64_t)(uintptr_t)(W + (size_t)brow1 * LDN_ + n0) + (uint64_t)bcb;
  const uint64_t dA = 16ull * 4ull;                 // advance 16 cols of A
  const uint64_t dB = (uint64_t)LDN_ * 16ull * 4ull; // advance 16 rows of W

  // LDS byte offsets (flat shared pointer truncates to LDS offset on AMDGPU)
  uint32_t ldsA[2], ldsB0[2], ldsB1[2];
#pragma unroll
  for (int b = 0; b < 2; ++b) {
    ldsA[b]  = (uint32_t)(uintptr_t)&sA[b][0] + (uint32_t)(arow * 80 + acb);
    ldsB0[b] = (uint32_t)(uintptr_t)&sB[b][0] + (uint32_t)(brow0 * 544 + bcb);
    ldsB1[b] = (uint32_t)(uintptr_t)&sB[b][0] + (uint32_t)(brow1 * 544 + bcb);
  }

  v8f acc00 = {}, acc01 = {}, acc10 = {}, acc11 = {};

  // prologue: stage K-tile 0 into buffer 0
  async_b128(ldsA[0], gA);
  async_b128(ldsB0[0], gB0);
  async_b128(ldsB1[0], gB1);
  gA += dA; gB0 += dB; gB1 += dB;
  wait_async0();
  __syncthreads();

  const int KT = K_ / 16;
  for (int kt = 0; kt < KT; ++kt) {
    const int cur = kt & 1, nxt = cur ^ 1;
    if (kt + 1 < KT) {   // stage next K-tile while WMMAs run on `cur`
      async_b128(ldsA[nxt], gA);
      async_b128(ldsB0[nxt], gB0);
      async_b128(ldsB1[nxt], gB1);
      gA += dA; gB0 += dB; gB1 += dB;
    }
#pragma unroll
    for (int kk = 0; kk < 16; kk += 4) {
      v2f a0 = *(const v2f*)(&sA[cur][(wm + l16) * 20 + kk + klo]);
      v2f a1 = *(const v2f*)(&sA[cur][(wm + 16 + l16) * 20 + kk + klo]);
      v2f b0, b1v;
      b0[0]  = sB[cur][(kk + klo) * 136 + wn + l16];
      b0[1]  = sB[cur][(kk + klo + 1) * 136 + wn + l16];
      b1v[0] = sB[cur][(kk + klo) * 136 + wn + 16 + l16];
      b1v[1] = sB[cur][(kk + klo + 1) * 136 + wn + 16 + l16];
      acc00 = wmma_f32(a0, b0,  acc00);
      acc01 = wmma_f32(a0, b1v, acc01);
      acc10 = wmma_f32(a1, b0,  acc10);
      acc11 = wmma_f32(a1, b1v, acc11);
    }
    wait_async0();     // my async writes into `nxt` are complete
    __syncthreads();   // everyone done: reads of `cur`, writes of `nxt`
  }

  // epilogue: +bias, activation / residual, scatter by token id
#pragma unroll
  for (int mi = 0; mi < 2; ++mi) {
#pragma unroll
    for (int ni = 0; ni < 2; ++ni) {
      v8f acc = (mi == 0) ? (ni == 0 ? acc00 : acc01)
                          : (ni == 0 ? acc10 : acc11);
      const int col = n0 + wn + ni * 16 + l16;
      const float bv = bias[col];
#pragma unroll
      for (int r = 0; r < 8; ++r) {
        const int row = wm + mi * 16 + half * 8 + r;
        if (m0 + row < cnt) {
          const int tok = sTok[row];
          if (GELU_) {
            Out[(size_t)tok * LDN_ + col] = gelu_exact(acc[r] + bv);
          } else {
            Out[(size_t)tok * LDN_ + col] =
                acc[r] + bv + Xres[(size_t)tok * LDN_ + col];
          }
        }
      }
    }
  }
}

// -------------------------------------------------------------- LayerNorm ---
__global__ __launch_bounds__(256) void ln_kernel(float* __restrict__ y,
                                                 const float* __restrict__ ln_w,
                                                 const float* __restrict__ ln_b) {
  const int tok = blockIdx.x;
  float* __restrict__ row = y + (size_t)tok * H_;
  const int t = threadIdx.x;

  float4 v = *(const float4*)(row + t * 4);
  float s  = v.x + v.y + v.z + v.w;
  float s2 = v.x * v.x + v.y * v.y + v.z * v.z + v.w * v.w;

  __shared__ float red0[256];
  __shared__ float red1[256];
  red0[t] = s; red1[t] = s2;
  __syncthreads();
#pragma unroll
  for (int off = 128; off > 0; off >>= 1) {
    if (t < off) { red0[t] += red0[t + off]; red1[t] += red1[t + off]; }
    __syncthreads();
  }
  const float mean = red0[0] * (1.0f / H_);
  const float var  = red1[0] * (1.0f / H_) - mean * mean;
  const float inv  = rsqrtf(var + 1e-12f);

  float4 w = *(const float4*)(ln_w + t * 4);
  float4 b = *(const float4*)(ln_b + t * 4);
  float4 o;
  o.x = w.x * (v.x - mean) * inv + b.x;
  o.y = w.y * (v.y - mean) * inv + b.y;
  o.z = w.z * (v.z - mean) * inv + b.z;
  o.w = w.w * (v.w - mean) * inv + b.w;
  *(float4*)(row + t * 4) = o;
}

// ----------------------------------------------------------------- launch ---
extern "C" void kernel_launch(void* const* d_in, const int* in_sizes, int n_in,
                              void* d_out, int out_size, void* d_ws, size_t ws_size,
                              hipStream_t stream) {
  (void)in_sizes; (void)n_in; (void)out_size; (void)ws_size;
  const float* X    = (const float*)d_in[0];
  const int*   eidx = (const int*)d_in[1];
  const float* W1   = (const float*)d_in[2];
  const float* b1   = (const float*)d_in[3];
  const float* W2   = (const float*)d_in[4];
  const float* b2   = (const float*)d_in[5];
  const float* lnw  = (const float*)d_in[6];
  const float* lnb  = (const float*)d_in[7];
  float* out = (float*)d_out;

  char* ws = (char*)d_ws;
  int*   counts  = (int*)ws;                                     //   32 B
  int*   buckets = (int*)(ws + 1024);                            //  256 KB
  float* interm  = (float*)(ws + 1024 + (size_t)E_ * NTOK * 4);  //  128 MB

  zero_counts_kernel<<<1, 32, 0, stream>>>(counts);
  route_kernel<<<NTOK / 256, 256, 0, stream>>>(eidx, counts, buckets);

  dim3 g1(I_ / 128, NTOK / 64, E_);   // worst-case grid, early-exit on counts
  moe_gemm<H_, H_, I_, true><<<g1, 256, 0, stream>>>(
      X, W1, b1, nullptr, counts, buckets, interm);

  dim3 g2(H_ / 128, NTOK / 64, E_);
  moe_gemm<I_, I_, H_, false><<<g2, 256, 0, stream>>>(
      interm, W2, b2, X, counts, buckets, out);

  ln_kernel<<<NTOK, 256, 0, stream>>>(out, lnw, lnb);
}